// OperatorBasisLayer_43069932044627
// MI455X (gfx1250) — compile-verified
//
#include <hip/hip_runtime.h>
#include <math.h>

#define DD 1024
#define BB 16
#define NEG_BIG (-1e30f)

typedef __attribute__((ext_vector_type(2))) float v2f;
typedef __attribute__((ext_vector_type(8))) float v8f;

// ---------------------------------------------------------------------------
// 0) softmax over beta (40 entries) -> w[40] in workspace
// ---------------------------------------------------------------------------
__global__ void k_softmax_beta(const float* __restrict__ beta, float* __restrict__ w) {
    if (threadIdx.x == 0) {
        float m = NEG_BIG;
        for (int k = 0; k < 40; ++k) m = fmaxf(m, beta[k]);
        float s = 0.f;
        for (int k = 0; k < 40; ++k) { float e = expf(beta[k] - m); w[k] = e; s += e; }
        float inv = 1.f / s;
        for (int k = 0; k < 40; ++k) w[k] *= inv;
    }
}

// ---------------------------------------------------------------------------
// 1) all pointwise / stencil ops (k = 0..31), writes out = x + 0.1*sum
// ---------------------------------------------------------------------------
__global__ void k_elementwise(const float* __restrict__ x, const float* __restrict__ w,
                              float* __restrict__ out) {
    __shared__ float sw[40];
    int t = threadIdx.x;
    if (t < 40) sw[t] = w[t];
    __syncthreads();

    int idx = blockIdx.x * blockDim.x + t;       // 0 .. B*D-1
    int b = idx >> 10;
    int d = idx & (DD - 1);
    const float* xr = x + b * DD;

    float c = xr[d];
    float l = xr[d > 0 ? d - 1 : 0];
    float r = xr[d < DD - 1 ? d + 1 : DD - 1];
    float lap = l - 2.f * c + r;

    float acc = 0.f;
    const float scales[4] = {0.5f, 1.f, 2.f, 4.f};
    #pragma unroll
    for (int i = 0; i < 4; ++i) acc += sw[i] * sinf(scales[i] * c);
    #pragma unroll
    for (int i = 0; i < 4; ++i) acc += sw[4 + i] * cosf(scales[i] * c);

    float c2 = c * c, c3 = c2 * c;
    float g = 0.5f * c * (1.f + tanhf(0.7978845608028654f * (c + 0.044715f * c3)));
    acc += sw[8] * g;
    acc += sw[9] * tanhf(c);
    acc += sw[10] * (1.f / (1.f + expf(-c)));
    acc += sw[11] * c2 + sw[12] * c3 + sw[13] * c2 * c2;

    float ax = fabsf(c);
    acc += sw[14] * (c / (1.f + 0.5f * ax + 1e-3f));
    acc += sw[15] * (c / (1.f + 1.0f * ax + 1e-3f));
    acc += sw[16] * (c / (1.f + 2.0f * ax + 1e-3f));

    const float dalpha[4] = {0.001f, 0.003f, 0.01f, 0.03f};
    #pragma unroll
    for (int i = 0; i < 4; ++i) acc += sw[17 + i] * (c + dalpha[i] * lap);

    // gaussian blurs (zero padding, matches lax.conv default)
    {
        const float sig[3] = {0.5f, 1.f, 2.f};
        const int   rad[3] = {2, 3, 6};
        #pragma unroll
        for (int s = 0; s < 3; ++s) {
            int rr = rad[s];
            float inv = 1.f / (sig[s] * sig[s]);
            float ksum = 0.f, o = 0.f;
            for (int dt = -rr; dt <= rr; ++dt) {
                float kv = expf(-0.5f * (float)(dt * dt) * inv);
                ksum += kv;
                int j = d + dt;
                float xv = (j >= 0 && j < DD) ? xr[j] : 0.f;
                o += kv * xv;
            }
            acc += sw[21 + s] * (o / ksum);
        }
    }
    // tempered LSE over {l,c,r} (replicate padding)
    {
        const float taus[4] = {0.5f, 1.f, 2.f, 4.f};
        float m = fmaxf(l, fmaxf(c, r));
        #pragma unroll
        for (int i = 0; i < 4; ++i) {
            float tt = taus[i], it = 1.f / tt;
            float sme = expf((l - m) * it) + expf((c - m) * it) + expf((r - m) * it);
            acc += sw[24 + i] * (m + tt * logf(sme));
        }
    }
    // neighbor softmax
    {
        const float taus[4] = {0.5f, 1.f, 2.f, 4.f};
        float dl = fabsf(l - c), dr = fabsf(r - c);
        #pragma unroll
        for (int i = 0; i < 4; ++i) {
            float it = 1.f / taus[i];
            float el = expf(-dl * it), er = expf(-dr * it);
            float sum = el + 1.f + er;
            acc += sw[28 + i] * ((el * l + c + er * r) / sum);
        }
    }
    out[idx] = c + 0.1f * acc;
}

// ---------------------------------------------------------------------------
// 2) RBF: d2[b][p] then Mc[b][p] = sum_sigma w_s * exp(-d2/(2s^2))
// ---------------------------------------------------------------------------
__global__ void k_rbf_phi(const float* __restrict__ x, const float* __restrict__ proto,
                          const float* __restrict__ w, float* __restrict__ Mc) {
    int lane = threadIdx.x & 31;
    int wid = (blockIdx.x * blockDim.x + threadIdx.x) >> 5;   // 0..511
    int b = wid >> 5;
    int p = wid & 31;
    const float* xr = x + b * DD;
    const float* pr = proto + p * DD;
    float s = 0.f;
    for (int d = lane; d < DD; d += 32) { float df = xr[d] - pr[d]; s += df * df; }
    #pragma unroll
    for (int off = 16; off; off >>= 1) s += __shfl_xor(s, off, 32);
    if (lane == 0) {
        float m = w[32] * expf(-s * 2.0f)       // sigma 0.5 : 1/(2*0.25)
                + w[33] * expf(-s * 0.5f)       // sigma 1.0 : 1/2
                + w[34] * expf(-s * 0.125f);    // sigma 2.0 : 1/8
        Mc[b * 32 + p] = m;
    }
}

// ---------------------------------------------------------------------------
// 3) WMMA f32 GEMM: out += 0.1 * (Mc[16x32] @ proj[32x1024]), one N-tile/wave
// ---------------------------------------------------------------------------
__global__ void k_rbf_gemm(const float* __restrict__ Mc, const float* __restrict__ proj,
                           float* __restrict__ out) {
    int lane = threadIdx.x;            // 0..31 (wave32)
    int n0 = blockIdx.x * 16;
    int half = lane >> 4, r = lane & 15;
    v8f c = {};
    #pragma unroll
    for (int k0 = 0; k0 < 32; k0 += 4) {
        int ka = k0 + 2 * half;
        v2f a, bf;
        a.x = Mc[r * 32 + ka];
        a.y = Mc[r * 32 + ka + 1];
        bf.x = proj[ka * DD + n0 + r];
        bf.y = proj[(ka + 1) * DD + n0 + r];
        c = __builtin_amdgcn_wmma_f32_16x16x4_f32(false, a, false, bf, (short)0, c,
                                                  false, false);
    }
    #pragma unroll
    for (int vr = 0; vr < 8; ++vr) {
        int M = vr + 8 * half;
        out[M * DD + n0 + r] += 0.1f * c[vr];
    }
}

// ---------------------------------------------------------------------------
// 4) WMMA f32 GEMM: out += 0.1*w39*(x[16x1024] @ W^T + bias)
// ---------------------------------------------------------------------------
__global__ void k_linear_gemm(const float* __restrict__ x, const float* __restrict__ W,
                              const float* __restrict__ bias, const float* __restrict__ w,
                              float* __restrict__ out) {
    int lane = threadIdx.x;
    int n0 = blockIdx.x * 16;
    int half = lane >> 4, r = lane & 15;
    float wl = w[39];
    const float* xrow = x + r * DD;             // A: M=r (batch)
    const float* wrow = W + (n0 + r) * DD;      // B[k][n] = W[n][k]
    v8f c = {};
    for (int k0 = 0; k0 < DD; k0 += 4) {
        int ka = k0 + 2 * half;
        v2f a  = *(const v2f*)(xrow + ka);
        v2f bf = *(const v2f*)(wrow + ka);
        c = __builtin_amdgcn_wmma_f32_16x16x4_f32(false, a, false, bf, (short)0, c,
                                                  false, false);
    }
    #pragma unroll
    for (int vr = 0; vr < 8; ++vr) {
        int M = vr + 8 * half;
        int n = n0 + r;
        out[M * DD + n] += 0.1f * wl * (c[vr] + bias[n]);
    }
}

// ---------------------------------------------------------------------------
// 5) rank-1 attention, both taus fused; one wave per (b,i), x row in LDS.
//    Row max of a*x_j is analytic (a*xmax or a*xmin) -> no online rescaling,
//    inner loop is a pure v_exp_f32 + fma stream.
// ---------------------------------------------------------------------------
__global__ void k_attn(const float* __restrict__ x, const float* __restrict__ wq,
                       const float* __restrict__ wk, const float* __restrict__ wv,
                       const float* __restrict__ wo, const float* __restrict__ w,
                       float* __restrict__ out) {
    __shared__ float xs[DD];
    int t = threadIdx.x;
    int b = blockIdx.x >> 7;                 // 128 blocks per batch row
    int i0 = (blockIdx.x & 127) * 8;
    const float* xr = x + b * DD;
    for (int j = t; j < DD; j += 256) xs[j] = xr[j];
    __syncthreads();

    float dqk = 0.f, dvo = 0.f;
    #pragma unroll
    for (int a = 0; a < 8; ++a) { dqk += wq[a] * wk[a]; dvo += wv[a] * wo[a]; }
    float base = dqk * 0.35355339059327373f;   // 1/sqrt(8)

    int wavei = t >> 5, lane = t & 31;
    int i = i0 + wavei;

    // per-wave min/max of the row (strided scan + shuffle reduce)
    float xmax = NEG_BIG, xmin = -NEG_BIG;
    for (int j = lane; j < DD; j += 32) {
        float xv = xs[j];
        xmax = fmaxf(xmax, xv);
        xmin = fminf(xmin, xv);
    }
    #pragma unroll
    for (int off = 16; off; off >>= 1) {
        xmax = fmaxf(xmax, __shfl_xor(xmax, off, 32));
        xmin = fminf(xmin, __shfl_xor(xmin, off, 32));
    }

    float xi = xs[i];
    float a0 = (base * 2.0f) * xi;             // tau = 0.5
    float a1 = base * xi;                      // tau = 1.0
    float m0 = fmaxf(a0 * xmax, a0 * xmin);    // analytic row max
    float m1 = fmaxf(a1 * xmax, a1 * xmin);

    float s0 = 0.f, t0 = 0.f, s1 = 0.f, t1 = 0.f;
    for (int j = lane; j < DD; j += 32) {
        float xv = xs[j];
        float e0 = __expf(a0 * xv - m0);
        s0 += e0; t0 += e0 * xv;
        float e1 = __expf(a1 * xv - m1);
        s1 += e1; t1 += e1 * xv;
    }
    #pragma unroll
    for (int off = 16; off; off >>= 1) {
        s0 += __shfl_xor(s0, off, 32);
        t0 += __shfl_xor(t0, off, 32);
        s1 += __shfl_xor(s1, off, 32);
        t1 += __shfl_xor(t1, off, 32);
    }
    if (lane == 0) {
        float r0 = dvo * t0 / s0;
        float r1 = dvo * t1 / s1;
        out[b * DD + i] += 0.1f * (w[35] * r0 + w[36] * r1);
    }
}

// ---------------------------------------------------------------------------
// 6) Sinkhorn streaming LSE pass: out[t] = -lse_s( -(x_t-x_s)^2/tau + p[s] ).
//    Arguments are bounded (diagonal term anchors the sum, potentials are
//    O(log D)), so no max subtraction needed: direct sum of exp, log at end.
// ---------------------------------------------------------------------------
__global__ void k_sink_lse(const float* __restrict__ x, const float* __restrict__ other,
                           float* __restrict__ outv, float invtau) {
    __shared__ float xs[DD];
    __shared__ float ps[DD];
    int t = threadIdx.x;
    int b = blockIdx.x >> 7;
    int i0 = (blockIdx.x & 127) * 8;
    const float* xr = x + b * DD;
    for (int j = t; j < DD; j += 256) {
        xs[j] = xr[j];
        ps[j] = other ? other[b * DD + j] : 0.f;
    }
    __syncthreads();
    int wavei = t >> 5, lane = t & 31;
    int i = i0 + wavei;
    float xi = xs[i];
    float s = 0.f;
    for (int j = lane; j < DD; j += 32) {
        float d = xi - xs[j];
        s += __expf(ps[j] - d * d * invtau);
    }
    #pragma unroll
    for (int off = 16; off; off >>= 1) s += __shfl_xor(s, off, 32);
    if (lane == 0) outv[b * DD + i] = -__logf(s);
}

// ---------------------------------------------------------------------------
// 7) Sinkhorn final: out[i] += 0.1*w_k * sum_j exp(k_ij + u_i + v_j) * x_j
// ---------------------------------------------------------------------------
__global__ void k_sink_acc(const float* __restrict__ x, const float* __restrict__ u,
                           const float* __restrict__ v, const float* __restrict__ w,
                           int widx, float invtau, float* __restrict__ out) {
    __shared__ float xs[DD];
    __shared__ float vs[DD];
    int t = threadIdx.x;
    int b = blockIdx.x >> 7;
    int i0 = (blockIdx.x & 127) * 8;
    const float* xr = x + b * DD;
    for (int j = t; j < DD; j += 256) { xs[j] = xr[j]; vs[j] = v[b * DD + j]; }
    __syncthreads();
    int wavei = t >> 5, lane = t & 31;
    int i = i0 + wavei;
    float xi = xs[i];
    float ui = u[b * DD + i];
    float acc = 0.f;
    for (int j = lane; j < DD; j += 32) {
        float d = xi - xs[j];
        acc += __expf(ui + vs[j] - d * d * invtau) * xs[j];
    }
    #pragma unroll
    for (int off = 16; off; off >>= 1) acc += __shfl_xor(acc, off, 32);
    if (lane == 0) out[b * DD + i] += 0.1f * w[widx] * acc;
}

// ---------------------------------------------------------------------------
// Launch
// ---------------------------------------------------------------------------
extern "C" void kernel_launch(void* const* d_in, const int* in_sizes, int n_in,
                              void* d_out, int out_size, void* d_ws, size_t ws_size,
                              hipStream_t stream) {
    const float* x     = (const float*)d_in[0];
    const float* beta  = (const float*)d_in[1];
    const float* W     = (const float*)d_in[2];
    const float* blin  = (const float*)d_in[3];
    const float* wq    = (const float*)d_in[4];
    const float* wk    = (const float*)d_in[5];
    const float* wv    = (const float*)d_in[6];
    const float* wo    = (const float*)d_in[7];
    const float* proto = (const float*)d_in[8];
    const float* proj  = (const float*)d_in[9];
    float* out = (float*)d_out;

    float* ws = (float*)d_ws;
    float* w  = ws;                // 40
    float* Mc = ws + 64;           // 512
    float* u  = ws + 1024;         // B*D
    float* v  = ws + 1024 + BB * DD;

    k_softmax_beta<<<1, 64, 0, stream>>>(beta, w);
    k_elementwise<<<(BB * DD) / 256, 256, 0, stream>>>(x, w, out);
    k_rbf_phi<<<64, 256, 0, stream>>>(x, proto, w, Mc);
    k_rbf_gemm<<<DD / 16, 32, 0, stream>>>(Mc, proj, out);
    k_attn<<<BB * (DD / 8), 256, 0, stream>>>(x, wq, wk, wv, wo, w, out);

    const float staus[2] = {0.5f, 1.0f};
    for (int ti = 0; ti < 2; ++ti) {
        float invt = 1.0f / staus[ti];
        for (int p = 0; p < 5; ++p) {
            k_sink_lse<<<BB * (DD / 8), 256, 0, stream>>>(x, (p == 0) ? nullptr : v, u, invt);
            k_sink_lse<<<BB * (DD / 8), 256, 0, stream>>>(x, u, v, invt);
        }
        k_sink_acc<<<BB * (DD / 8), 256, 0, stream>>>(x, u, v, w, 37 + ti, invt, out);
    }

    k_linear_gemm<<<DD / 16, 32, 0, stream>>>(x, W, blin, w, out);
}